// CombinedLoss_8486855377153
// MI455X (gfx1250) — compile-verified
//
#include <hip/hip_runtime.h>
#include <hip/hip_bf16.h>

typedef unsigned int u32;
typedef unsigned long long u64;
typedef __attribute__((ext_vector_type(8))) u32   v8u;  // 8 dwords = 16 packed f16
typedef __attribute__((ext_vector_type(8))) float v8f;

#define SAMPLES 32
#define N_PER   262144            // 512*512 per sample (C=1)
#define TOTAL   (SAMPLES * N_PER) // 8,388,608
#define RADIX   256
#define EPT     16                // elements per thread in radix kernels
#define BLK_T   256               // threads per block
#define EPB     (EPT * BLK_T)     // 4096 elements per radix block
#define BLOCKS_PER_SAMPLE (N_PER / EPB)   // 64
#define RADIX_BLOCKS (TOTAL / EPB)        // 2048
#define TILES_PER_SAMPLE 1024     // 256-element tiles
#define TOTAL_TILES (SAMPLES * TILES_PER_SAMPLE) // 32768
#define LOSS_EPS 1e-6f
#define F16_ONE 0x3C00u           // bit pattern of _Float16 1.0

// ---------------------------------------------------------------------------
// Init: zero the scalar accumulators (harness poisons ws, and launches must be
// deterministic, so every call re-zeros).
// ---------------------------------------------------------------------------
__global__ void init_acc(float* acc /* 129 floats: bce,inter[32],s[32],t[32],lov[32] */) {
  int tid = threadIdx.x;
  if (tid < 129) acc[tid] = 0.0f;
}

// ---------------------------------------------------------------------------
// Pass 1 (fused, streaming): BCE terms, per-sample sigmoid/target/intersection
// sums, hinge errors -> descending-sortable u64 keys (key<<32 | label).
// 1024 elements per block; 256 blocks per sample (blocks never straddle).
// ---------------------------------------------------------------------------
__global__ void fused_pre(const float* __restrict__ x, const float* __restrict__ t,
                          u64* __restrict__ keys,
                          float* bceSum, float* interS, float* sS, float* tS) {
  __shared__ float red[256];
  const int tid = threadIdx.x;
  const size_t blockStart = (size_t)blockIdx.x * 1024;
  const int sample = blockIdx.x >> 8;

  float lb = 0.f, li = 0.f, ls = 0.f, lt = 0.f;
#pragma unroll
  for (int i = 0; i < 4; ++i) {
    size_t idx = blockStart + (size_t)i * 256 + tid;
    float xv = x[idx], tv = t[idx];
    float ax = fabsf(xv);
    lb += fmaxf(xv, 0.f) - xv * tv + log1pf(expf(-ax));
    float s = 1.f / (1.f + expf(-xv));
    li += s * tv; ls += s; lt += tv;
    // Lovasz hinge error and descending-order radix key
    float sign = 2.f * tv - 1.f;
    float err = 1.f - xv * sign;
    u32 u = __float_as_uint(err);
    u32 asc = (u & 0x80000000u) ? ~u : (u | 0x80000000u); // ascending-sortable
    u32 key = ~asc;                                        // descending
    keys[idx] = ((u64)key << 32) | (u64)((tv != 0.f) ? 1u : 0u);
  }

  auto blockReduce = [&](float v) -> float {
    red[tid] = v; __syncthreads();
    for (int off = 128; off > 0; off >>= 1) {
      if (tid < off) red[tid] += red[tid + off];
      __syncthreads();
    }
    float r = red[0]; __syncthreads();
    return r;
  };
  float rb = blockReduce(lb);
  float ri = blockReduce(li);
  float rs = blockReduce(ls);
  float rt = blockReduce(lt);
  if (tid == 0) {
    atomicAdd(bceSum, rb);
    atomicAdd(&interS[sample], ri);
    atomicAdd(&sS[sample], rs);
    atomicAdd(&tS[sample], rt);
  }
}

// ---------------------------------------------------------------------------
// Radix histogram pass: stage 4096 u64 into LDS with CDNA5 async global->LDS
// copies, then digit-count. hist layout: [sample][digit][blockInSample].
// ---------------------------------------------------------------------------
__global__ void radix_hist(const u64* __restrict__ src, u32* __restrict__ hist, int shift) {
  __shared__ u64 stage[EPB];          // 32 KB of the 320 KB WGP LDS
  __shared__ u32 cnt[RADIX];
  const int tid = threadIdx.x;
  const int blk = blockIdx.x;
  const int sample = blk >> 6;
  const int blkInS = blk & 63;
  const size_t blockStart = (size_t)blk * EPB;

  // Async staging: one b64 per lane per issue; 16 issues cover the block tile.
#pragma unroll
  for (int i = 0; i < EPT; ++i) {
    unsigned ldsOff = (unsigned)(size_t)&stage[i * 256 + tid];
    unsigned long long ga = (unsigned long long)(src + blockStart + (size_t)i * 256 + tid);
    asm volatile("global_load_async_to_lds_b64 %0, %1, off"
                 :: "v"(ldsOff), "v"(ga) : "memory");
  }
  cnt[tid] = 0;
  asm volatile("s_wait_asynccnt 0x0" ::: "memory");
  __syncthreads();

#pragma unroll
  for (int i = 0; i < EPT; ++i) {
    u32 dig = (u32)(stage[i * 256 + tid] >> (32 + shift)) & 255u;
    atomicAdd(&cnt[dig], 1u);
  }
  __syncthreads();
  hist[((size_t)(sample * RADIX + tid)) * BLOCKS_PER_SAMPLE + blkInS] = cnt[tid];
}

// ---------------------------------------------------------------------------
// Exclusive scan of one sample's 256x64 histogram (digit-major flatten), in
// place. One block per sample; thread tid owns digit tid's 64 block counts.
// ---------------------------------------------------------------------------
__global__ void radix_scan(u32* __restrict__ hist) {
  __shared__ u32 tsum[256];
  const int s = blockIdx.x, tid = threadIdx.x;
  u32* h = hist + (size_t)s * RADIX * BLOCKS_PER_SAMPLE;
  const int base0 = tid * BLOCKS_PER_SAMPLE;
  u32 local = 0;
#pragma unroll 4
  for (int i = 0; i < BLOCKS_PER_SAMPLE; ++i) local += h[base0 + i];
  tsum[tid] = local; __syncthreads();
  for (int off = 1; off < 256; off <<= 1) {
    u32 v = (tid >= off) ? tsum[tid - off] : 0u;
    __syncthreads();
    tsum[tid] += v;
    __syncthreads();
  }
  u32 run = (tid == 0) ? 0u : tsum[tid - 1];
#pragma unroll 4
  for (int i = 0; i < BLOCKS_PER_SAMPLE; ++i) { u32 v = h[base0 + i]; h[base0 + i] = run; run += v; }
}

// ---------------------------------------------------------------------------
// Stable radix scatter: per-wave ballot multisplit ranks (wave32), per-block
// wave-count scan in LDS, per-digit running bases seeded from scanned hist.
// ---------------------------------------------------------------------------
__global__ void radix_scatter(const u64* __restrict__ src, u64* __restrict__ dst,
                              const u32* __restrict__ hist, int shift) {
  __shared__ u32 waveCnt[8][RADIX];
  __shared__ u32 running[RADIX];
  const int tid = threadIdx.x;
  const int lane = tid & 31, wave = tid >> 5;
  const int blk = blockIdx.x;
  const int sample = blk >> 6, blkInS = blk & 63;
  const size_t sampleBase = (size_t)sample * N_PER;
  const size_t blockStart = (size_t)blk * EPB;

  running[tid] = hist[((size_t)(sample * RADIX + tid)) * BLOCKS_PER_SAMPLE + blkInS];
  __syncthreads();

  for (int i = 0; i < EPT; ++i) {
#pragma unroll
    for (int w = 0; w < 8; ++w) waveCnt[w][tid] = 0;
    __syncthreads();

    u64 kv = src[blockStart + (size_t)i * 256 + tid];
    u32 dig = (u32)(kv >> (32 + shift)) & 255u;
    u32 match = 0xffffffffu;
#pragma unroll
    for (int b = 0; b < 8; ++b) {
      u32 bal = (u32)__ballot((dig >> b) & 1u);
      match &= ((dig >> b) & 1u) ? bal : ~bal;
    }
    u32 rank = __popc(match & ((1u << lane) - 1u));
    if (rank == 0) waveCnt[wave][dig] = __popc(match);
    __syncthreads();

    u32 off = running[dig];
#pragma unroll
    for (int w = 0; w < 8; ++w)
      if (w < wave) off += waveCnt[w][dig];
    dst[sampleBase + off + rank] = kv;
    __syncthreads();

    u32 add = 0;
#pragma unroll
    for (int w = 0; w < 8; ++w) add += waveCnt[w][tid];
    running[tid] += add;
    __syncthreads();
  }
}

// ---------------------------------------------------------------------------
// Per-256-tile label sums of the sorted stream (for the hierarchical scan).
// ---------------------------------------------------------------------------
__global__ void tile_sums(const u64* __restrict__ keys, float* __restrict__ tsums) {
  __shared__ float red[256];
  const int tile = blockIdx.x, tid = threadIdx.x;
  float v = (float)(u32)(keys[(size_t)tile * 256 + tid] & 1ull);
  red[tid] = v; __syncthreads();
  for (int off = 128; off > 0; off >>= 1) {
    if (tid < off) red[tid] += red[tid + off];
    __syncthreads();
  }
  if (tid == 0) tsums[tile] = red[0];
}

// Exclusive scan of 1024 tile sums per sample; also emits p = total positives.
__global__ void tile_scan(const float* __restrict__ tsums, float* __restrict__ toff,
                          float* __restrict__ gts) {
  __shared__ float tl[256];
  const int s = blockIdx.x, tid = threadIdx.x;
  const float* in = tsums + (size_t)s * TILES_PER_SAMPLE;
  float* out = toff + (size_t)s * TILES_PER_SAMPLE;
  float vals[4]; float loc = 0.f;
#pragma unroll
  for (int i = 0; i < 4; ++i) { vals[i] = in[tid * 4 + i]; loc += vals[i]; }
  tl[tid] = loc; __syncthreads();
  for (int off = 1; off < 256; off <<= 1) {
    float v = (tid >= off) ? tl[tid - off] : 0.f;
    __syncthreads();
    tl[tid] += v;
    __syncthreads();
  }
  float run = (tid == 0) ? 0.f : tl[tid - 1];
#pragma unroll
  for (int i = 0; i < 4; ++i) { out[tid * 4 + i] = run; run += vals[i]; }
  if (tid == 255) gts[s] = run;
}

// ---------------------------------------------------------------------------
// Lovasz evaluation over sorted keys. Each wave handles one 256-element tile:
// prefixes of each chunk's first 8 labels via ONE v_wmma_f32_16x16x32_f16 (D = L * X),
// 16-wide chunk-offset scan in LDS, then per-element Jaccard grad * relu(err).
//
// Backend-bug hardening for this clang-23-rc2 toolchain:
//  * WMMA emitted as raw inline asm (the llvm.amdgcn.wmma intrinsics crash
//    ISel for gfx1250).
//  * f16 operands synthesized as packed integer dwords (0x3C00 == 1.0h): any
//    f16-typed IR crashes 16-bit subregister codegen.
//  * EVERY element of the 256-bit asm operands is derived from an opaque
//    VGPR zero (v_mov_b32 via asm) so the REG_SEQUENCEs are pure-VGPR and no
//    wide SGPR->VGPR tuple copies exist (SIFixSGPRCopies segfaults on those).
// ---------------------------------------------------------------------------
__global__ void lovasz_eval(const u64* __restrict__ keys, const float* __restrict__ tileOff,
                            const float* __restrict__ gts, float* __restrict__ lovOut) {
  __shared__ float chS[8][16];
  __shared__ float red[256];
  const int tid = threadIdx.x;
  const int lane = tid & 31, wave = tid >> 5;
  const int globalTile = blockIdx.x * 8 + wave;      // 8 tiles/block, same sample
  const int sample = globalTile >> 10;
  const int tileInS = globalTile & 1023;
  const u64* base = keys + (size_t)sample * N_PER + (size_t)tileInS * 256;

  const int c = lane & 15;    // chunk (== row index M of A, == column N of B/D)
  const int g = lane >> 4;    // lane half

  // Opaque VGPR-resident zero: keeps all vector elements in VALU registers.
  u32 vzero;
  asm volatile("v_mov_b32 %0, 0" : "=v"(vzero));

  // A = 16x16 lower-triangular ones, zero-padded to 16x32 (K>=16 zero).
  // f16 A layout: lane half g supplies K = g*8 + h for halves h=0..7; halves
  // 8..15 are K>=16 (zero). VGPR v packs halves 2v (lo) and 2v+1 (hi).
  v8u a;
#pragma unroll
  for (int v = 0; v < 4; ++v) {
    int K0 = g * 8 + 2 * v;
    int K1 = K0 + 1;
    u32 lo = (K0 <= c) ? F16_ONE : vzero;
    u32 hi = (K1 <= c) ? F16_ONE : vzero;
    a[v] = lo | (hi << 16);
  }
#pragma unroll
  for (int v = 4; v < 8; ++v) a[v] = vzero;

  // B = X padded to 32x16: lanes 0-15 of column 'lane' pair halves 0-7 with K=0..7, 8-15 with K=16..23 (labels
  // of elements lane*16 + h); lanes 16-31 (K=8..15, 24..31) carry zeros.
  v8u b;
#pragma unroll
  for (int v = 0; v < 8; ++v) {
    u32 lo = vzero, hi = vzero;
    if (lane < 16) {
      lo = ((u32)(base[lane * 16 + 2 * v] & 1ull)) ? F16_ONE : vzero;
      hi = ((u32)(base[lane * 16 + 2 * v + 1] & 1ull)) ? F16_ONE : vzero;
    }
    b[v] = lo | (hi << 16);
  }

  // Accumulator: splat of the opaque VGPR zero (no SGPR->VReg_256 copy).
  const float fz = __uint_as_float(vzero);
  v8f d;
#pragma unroll
  for (int j = 0; j < 8; ++j) d[j] = fz;

  // D = A*B + C with C tied to D (accumulate in place). Four trailing v_nops
  // cover the CDNA5 WMMA_*F16 -> VALU RAW hazard (ISA 7.12.1) since the
  // compiler cannot see through the asm.
  asm volatile(
      "v_wmma_f32_16x16x32_f16 %0, %1, %2, %0\n\t"
      "v_nop\n\t"
      "v_nop\n\t"
      "v_nop\n\t"
      "v_nop"
      : "+v"(d)
      : "v"(a), "v"(b));
  // D[m][n] = sum_{e<=min(m,7)} of chunk n's label e; labels 8..15 hit A's zero K=16..23.
  // Lane layout: VGPR r -> M = r (lanes 0-15) / 8+r (lanes 16-31), N = lane&15.

  if (lane >= 16) chS[wave][lane - 16] = d[7];   // M=15 row = sum of chunk's first 8 labels
  __syncthreads();
  float coff = 0.f;
#pragma unroll
  for (int j = 0; j < 16; ++j)
    if (j < c) coff += chS[wave][j];             // exclusive chunk offset

  const float toff = tileOff[sample * TILES_PER_SAMPLE + tileInS];
  const float p = gts[sample];

  float acc = 0.f;
  const int eBase = g * 8;
#pragma unroll
  for (int j = 0; j < 8; ++j) {
    int e = eBase + j;
    u64 kv = base[c * 16 + e];
    float y = (float)(u32)(kv & 1ull);
    // invert descending key -> error float
    u32 ascv = ~(u32)(kv >> 32);
    u32 ubits = (ascv & 0x80000000u) ? (ascv & 0x7fffffffu) : ~ascv;
    float err = __uint_as_float(ubits);
    float relu = err > 0.f ? err : 0.f;
    float P = toff + coff + d[j];
    float kf = (float)(tileInS * 256 + c * 16 + e + 1);    // 1-indexed position
    float Jk = 1.f - (p - P) / (p + kf - P);
    float km = kf - 1.f, Pm = P - y;
    float Jm = (km == 0.f) ? 0.f : 1.f - (p - Pm) / (p + km - Pm);
    acc += relu * (Jk - Jm);
  }

  red[tid] = acc; __syncthreads();
  for (int off = 128; off > 0; off >>= 1) {
    if (tid < off) red[tid] += red[tid + off];
    __syncthreads();
  }
  if (tid == 0) atomicAdd(&lovOut[sample], red[0]);
}

// ---------------------------------------------------------------------------
// Final combine: out = 0.5*(bce + dice) + 0.5*lovasz
// ---------------------------------------------------------------------------
__global__ void combine(const float* acc, float* out) {
  __shared__ float rd[32], rl[32];
  const int tid = threadIdx.x;  // 32 threads
  const float* bceSum = acc;
  const float* interS = acc + 1;
  const float* sS = acc + 33;
  const float* tS = acc + 65;
  const float* lovS = acc + 97;
  float dice = 1.f - (2.f * interS[tid] + LOSS_EPS) / (sS[tid] + tS[tid] + LOSS_EPS);
  rd[tid] = dice; rl[tid] = lovS[tid]; __syncthreads();
  for (int off = 16; off > 0; off >>= 1) {
    if (tid < off) { rd[tid] += rd[tid + off]; rl[tid] += rl[tid + off]; }
    __syncthreads();
  }
  if (tid == 0) {
    float bce = bceSum[0] / (float)TOTAL;
    out[0] = 0.5f * (bce + rd[0] / 32.f) + 0.5f * (rl[0] / 32.f);
  }
}

// ---------------------------------------------------------------------------
extern "C" void kernel_launch(void* const* d_in, const int* in_sizes, int n_in,
                              void* d_out, int out_size, void* d_ws, size_t ws_size,
                              hipStream_t stream) {
  (void)in_sizes; (void)n_in; (void)out_size; (void)ws_size;
  const float* x = (const float*)d_in[0];
  const float* t = (const float*)d_in[1];
  float* out = (float*)d_out;

  // Workspace layout (8-byte types first for alignment)
  u64* keysA = (u64*)d_ws;                              // TOTAL u64
  u64* keysB = keysA + TOTAL;                           // TOTAL u64
  u32* hist = (u32*)(keysB + TOTAL);                    // 32*256*64 u32
  float* tsums = (float*)(hist + SAMPLES * RADIX * BLOCKS_PER_SAMPLE); // 32768
  float* toff = tsums + TOTAL_TILES;                    // 32768
  float* gts = toff + TOTAL_TILES;                      // 32
  float* acc = gts + SAMPLES;                           // 129: bce,inter,s,t,lov

  init_acc<<<1, 256, 0, stream>>>(acc);
  fused_pre<<<TOTAL / 1024, 256, 0, stream>>>(x, t, keysA, acc, acc + 1, acc + 33, acc + 65);

  for (int pass = 0; pass < 4; ++pass) {
    u64* s = (pass & 1) ? keysB : keysA;
    u64* d = (pass & 1) ? keysA : keysB;
    int shift = pass * 8;
    radix_hist<<<RADIX_BLOCKS, 256, 0, stream>>>(s, hist, shift);
    radix_scan<<<SAMPLES, 256, 0, stream>>>(hist);
    radix_scatter<<<RADIX_BLOCKS, 256, 0, stream>>>(s, d, hist, shift);
  }
  // 4 passes: A->B->A->B->A ; sorted data ends in keysA

  tile_sums<<<TOTAL_TILES, 256, 0, stream>>>(keysA, tsums);
  tile_scan<<<SAMPLES, 256, 0, stream>>>(tsums, toff, gts);
  lovasz_eval<<<TOTAL_TILES / 8, 256, 0, stream>>>(keysA, toff, gts, acc + 97);
  combine<<<1, 32, 0, stream>>>(acc, out);
}